// DijkstraPQ_22162031247489
// MI455X (gfx1250) — compile-verified
//
#include <hip/hip_runtime.h>
#include <stdint.h>

// Problem constants (B=2, H=8, N=512)
#define GG   16      // B*H independent graphs
#define NN   512     // matrix dimension
#define TILE 64      // FW block size
#define NT   8       // NN / TILE
#define LPAD 68      // padded LDS row stride in floats (272B, 16B aligned)

// ---------------------------------------------------------------------------
// CDNA5 async global->LDS tile staging (gfx1250): each lane issues b128
// async copies; completion tracked with ASYNCcnt.
// ---------------------------------------------------------------------------
__device__ __forceinline__ void async_tile_load(const float* __restrict__ D,
                                                int g, int ti, int tj,
                                                float (*S)[LPAD], int t) {
  // tile base: D + g*N*N + (ti*64)*N + tj*64
  const float* base = D + ((size_t)g << 18) + ((size_t)ti << 15) + ((size_t)tj << 6);
  uint32_t s0 = (uint32_t)(uintptr_t)(&S[0][0]);  // low 32 bits == LDS offset
#pragma unroll
  for (int q = 0; q < 4; ++q) {
    int chunk = t + q * 256;          // 1024 chunks of 16B cover the 16KB tile
    int row   = chunk >> 4;           // 0..63
    int seg   = chunk & 15;           // 16B segment within row
    uint32_t loff  = s0 + (uint32_t)(row * (LPAD * 4) + seg * 16);
    uint64_t gaddr = (uint64_t)(uintptr_t)(base + row * NN + seg * 4);
    asm volatile("global_load_async_to_lds_b128 %0, %1, off"
                 :: "v"(loff), "v"(gaddr) : "memory");
  }
}

__device__ __forceinline__ void async_wait_all() {
  asm volatile("s_wait_asynccnt 0" ::: "memory");
}

__device__ __forceinline__ void store_tile(float* __restrict__ D,
                                           int g, int ti, int tj,
                                           const float (*S)[LPAD], int t) {
  int r0 = (t >> 4) << 2, c0 = (t & 15) << 2;
  float* base = D + ((size_t)g << 18) + ((size_t)ti << 15) + ((size_t)tj << 6)
                  + (size_t)r0 * NN + c0;
#pragma unroll
  for (int ri = 0; ri < 4; ++ri)
    *(float4*)(base + ri * NN) = *(const float4*)&S[r0 + ri][c0];
}

// ---------------------------------------------------------------------------
// D = (adj == 0 ? +inf : adj); D[i][i] = 0
// ---------------------------------------------------------------------------
__global__ void __launch_bounds__(256) fw_init(const float* __restrict__ adj,
                                               float* __restrict__ D) {
  size_t idx = ((size_t)blockIdx.x * 256 + threadIdx.x) * 4;
  float4 v = *(const float4*)(adj + idx);
  int rem = (int)(idx & (size_t)(NN * NN - 1));
  int i = rem >> 9;
  int j = rem & (NN - 1);
  const float inf = __builtin_inff();
  float4 o;
  o.x = (i == j    ) ? 0.f : ((v.x == 0.f) ? inf : v.x);
  o.y = (i == j + 1) ? 0.f : ((v.y == 0.f) ? inf : v.y);
  o.z = (i == j + 2) ? 0.f : ((v.z == 0.f) ? inf : v.z);
  o.w = (i == j + 3) ? 0.f : ((v.w == 0.f) ? inf : v.w);
  *(float4*)(D + idx) = o;
}

// ---------------------------------------------------------------------------
// Fused phases 1+2. Every block:
//   (a) loads the diagonal tile (k,k) and its own panel tile,
//   (b) runs the diagonal self-FW locally in LDS (redundant across blocks,
//       but removes the 16-block phase-1 launch / global round-trip),
//   (c) runs the panel update against the locally-finalized diagonal.
// Block m==0 (per graph) also writes the finalized diagonal tile back.
// ---------------------------------------------------------------------------
__global__ void __launch_bounds__(256) fw_phase12(float* __restrict__ D, int k) {
  __shared__ __align__(16) float Ts[TILE][LPAD];  // target panel tile
  __shared__ __align__(16) float Ks[TILE][LPAD];  // diagonal tile
  const int g = blockIdx.y;
  const int m = blockIdx.x;           // 0..13
  const int t = threadIdx.x;
  const bool isRow = (m < NT - 1);
  int mm = isRow ? m : m - (NT - 1);
  int o  = (mm < k) ? mm : mm + 1;    // skip the diagonal tile index
  int ti = isRow ? k : o;
  int tj = isRow ? o : k;
  async_tile_load(D, g, k, k, Ks, t);
  async_tile_load(D, g, ti, tj, Ts, t);
  async_wait_all();
  __syncthreads();
  const int r0 = (t >> 4) << 2, c0 = (t & 15) << 2;

  // --- phase 1: FW on the diagonal tile, locally in LDS ---
  for (int kk = 0; kk < TILE; ++kk) {
    float a[4], b[4];
#pragma unroll
    for (int ri = 0; ri < 4; ++ri) a[ri] = Ks[r0 + ri][kk];
    float4 bv = *(const float4*)&Ks[kk][c0];
    b[0] = bv.x; b[1] = bv.y; b[2] = bv.z; b[3] = bv.w;
#pragma unroll
    for (int ri = 0; ri < 4; ++ri)
#pragma unroll
      for (int ci = 0; ci < 4; ++ci)
        Ks[r0 + ri][c0 + ci] = fminf(Ks[r0 + ri][c0 + ci], a[ri] + b[ci]);
    __syncthreads();
  }

  // --- phase 2: panel update against finalized diagonal ---
  for (int kk = 0; kk < TILE; ++kk) {
    float a[4], b[4];
    if (isRow) {                       // tile (k,o): C = min(C, Dk[i][kk] + C[kk][j])
#pragma unroll
      for (int ri = 0; ri < 4; ++ri) a[ri] = Ks[r0 + ri][kk];
      float4 bv = *(const float4*)&Ts[kk][c0];
      b[0] = bv.x; b[1] = bv.y; b[2] = bv.z; b[3] = bv.w;
    } else {                           // tile (o,k): C = min(C, C[i][kk] + Dk[kk][j])
#pragma unroll
      for (int ri = 0; ri < 4; ++ri) a[ri] = Ts[r0 + ri][kk];
      float4 bv = *(const float4*)&Ks[kk][c0];
      b[0] = bv.x; b[1] = bv.y; b[2] = bv.z; b[3] = bv.w;
    }
#pragma unroll
    for (int ri = 0; ri < 4; ++ri)
#pragma unroll
      for (int ci = 0; ci < 4; ++ci)
        Ts[r0 + ri][c0 + ci] = fminf(Ts[r0 + ri][c0 + ci], a[ri] + b[ci]);
    __syncthreads();
  }

  store_tile(D, g, ti, tj, Ts, t);
  if (m == 0) store_tile(D, g, k, k, Ks, t);  // one block writes the diag back
}

// ---------------------------------------------------------------------------
// Phase 3 (bulk): C(ti,tj) = min(C, A(ti,k) (minplus) B(k,tj)).
// A,B async-staged to LDS; C in registers; barrier-free kk loop.
// ---------------------------------------------------------------------------
__global__ void __launch_bounds__(256) fw_phase3(float* __restrict__ D, int k) {
  __shared__ __align__(16) float As[TILE][LPAD];
  __shared__ __align__(16) float Bs[TILE][LPAD];
  const int g = blockIdx.y;
  const int m = blockIdx.x;           // 0..48
  const int t = threadIdx.x;
  int mi = m / (NT - 1), mj = m % (NT - 1);
  int ti = (mi < k) ? mi : mi + 1;
  int tj = (mj < k) ? mj : mj + 1;
  async_tile_load(D, g, ti, k, As, t);
  async_tile_load(D, g, k, tj, Bs, t);

  const int r0 = (t >> 4) << 2, c0 = (t & 15) << 2;
  float* cbase = D + ((size_t)g << 18) + ((size_t)ti << 15) + ((size_t)tj << 6)
                   + (size_t)r0 * NN + c0;
  float c[4][4];
#pragma unroll
  for (int ri = 0; ri < 4; ++ri) {        // overlaps with async LDS fills
    float4 v = *(const float4*)(cbase + ri * NN);
    c[ri][0] = v.x; c[ri][1] = v.y; c[ri][2] = v.z; c[ri][3] = v.w;
  }

  async_wait_all();
  __syncthreads();

#pragma unroll 8
  for (int kk = 0; kk < TILE; ++kk) {
    float a[4], b[4];
#pragma unroll
    for (int ri = 0; ri < 4; ++ri) a[ri] = As[r0 + ri][kk];
    float4 bv = *(const float4*)&Bs[kk][c0];
    b[0] = bv.x; b[1] = bv.y; b[2] = bv.z; b[3] = bv.w;
#pragma unroll
    for (int ri = 0; ri < 4; ++ri)
#pragma unroll
      for (int ci = 0; ci < 4; ++ci)
        c[ri][ci] = fminf(c[ri][ci], a[ri] + b[ci]);
  }

#pragma unroll
  for (int ri = 0; ri < 4; ++ri) {
    float4 v; v.x = c[ri][0]; v.y = c[ri][1]; v.z = c[ri][2]; v.w = c[ri][3];
    *(float4*)(cbase + ri * NN) = v;
  }
}

// ---------------------------------------------------------------------------
extern "C" void kernel_launch(void* const* d_in, const int* in_sizes, int n_in,
                              void* d_out, int out_size, void* d_ws, size_t ws_size,
                              hipStream_t stream) {
  const float* adj = (const float*)d_in[0];
  float* D = (float*)d_out;

  // 16*512*512 floats, 4 per thread, 256 threads/block -> 4096 blocks
  fw_init<<<4096, 256, 0, stream>>>(adj, D);

  for (int k = 0; k < NT; ++k) {
    fw_phase12<<<dim3(2 * (NT - 1), GG), 256, 0, stream>>>(D, k);
    fw_phase3<<<dim3((NT - 1) * (NT - 1), GG), 256, 0, stream>>>(D, k);
  }

  (void)in_sizes; (void)n_in; (void)out_size; (void)d_ws; (void)ws_size;
}